// Model_3470333575378
// MI455X (gfx1250) — compile-verified
//
#include <hip/hip_runtime.h>

// merge_attn_states for MI455X (gfx1250, wave32).
// Memory-bound: 768 MiB streaming traffic -> ~35us floor at 23.3 TB/s.
// Strategy: one wave32 handles one 128-float head row per iteration
// (32 lanes x float4 = one global_load_b128), ILP=4 rows per wave,
// non-temporal hints since working set (768MB) >> 192MB L2.
// LSE addresses are wave-uniform: readfirstlane forces the compiler to
// prove it and emit s_load_b32 (scalar path / constant cache), keeping
// the vector memory pipe 100% for the streaming B128 traffic.

constexpr int NUM_TOKENS = 16384;
constexpr int NUM_HEADS  = 32;   // power of two -> shift/mask indexing
constexpr int HEAD_SIZE  = 128;  // 32 lanes * 4 floats
constexpr int ILP        = 4;    // (token,head) pairs per wave
constexpr int WAVES_PER_BLOCK = 8;

typedef float vfloat4 __attribute__((ext_vector_type(4)));

__global__ __launch_bounds__(256) void merge_attn_states_kernel(
    const float* __restrict__ p_out,   // [tokens, heads, 128]
    const float* __restrict__ p_lse,   // [heads, tokens]
    const float* __restrict__ s_out,   // [tokens, heads, 128]
    const float* __restrict__ s_lse,   // [heads, tokens]
    float* __restrict__ out,           // [tokens, heads, 128]
    float* __restrict__ out_lse)       // [heads, tokens]
{
    const int lane = threadIdx.x & 31;

    // Wave index: provably wave-uniform -> SGPR arithmetic downstream.
    const int wid   = __builtin_amdgcn_readfirstlane(threadIdx.x >> 5);
    const int waveg = blockIdx.x * WAVES_PER_BLOCK + wid;
    const int pair0 = waveg * ILP;    // pair = token*NUM_HEADS + head

    const float INF = __builtin_inff();

    // ---- LSE loads: uniform address -> s_load_b32 via constant cache ----
    float pl[ILP], sl[ILP];
    int   lidx[ILP];
#pragma unroll
    for (int i = 0; i < ILP; ++i) {
        const int pair = pair0 + i;
        lidx[i] = (pair & (NUM_HEADS - 1)) * NUM_TOKENS + (pair >> 5); // [head, token]
        pl[i] = p_lse[lidx[i]];
        sl[i] = s_lse[lidx[i]];
    }

    // ---- big streaming loads (global_load_b128, non-temporal) ----
    vfloat4 pv[ILP], sv[ILP];
#pragma unroll
    for (int i = 0; i < ILP; ++i) {
        const size_t base = ((size_t)(pair0 + i) << 7) + ((size_t)lane << 2);
        pv[i] = __builtin_nontemporal_load((const vfloat4*)(p_out + base));
        sv[i] = __builtin_nontemporal_load((const vfloat4*)(s_out + base));
    }

    // ---- scale math (hidden under the loads above) ----
    float pscale[ILP], sscale[ILP];
#pragma unroll
    for (int i = 0; i < ILP; ++i) {
        float p = pl[i];
        float s = sl[i];
        // vLLM sanitization: +inf -> -inf
        if (p == INF) p = -INF;
        if (s == INF) s = -INF;
        const float m   = fmaxf(p, s);
        const float pe  = expf(p - m);
        const float se  = expf(s - m);
        const float sum = pe + se;
        const float inv = 1.0f / sum;
        pscale[i] = pe * inv;
        sscale[i] = se * inv;
        if (lane == 0) {
            out_lse[lidx[i]] = logf(sum) + m;
        }
    }

    // ---- blend + streaming store (global_store_b128, non-temporal) ----
#pragma unroll
    for (int i = 0; i < ILP; ++i) {
        const size_t base = ((size_t)(pair0 + i) << 7) + ((size_t)lane << 2);
        vfloat4 o;
        o.x = fmaf(pv[i].x, pscale[i], sv[i].x * sscale[i]);
        o.y = fmaf(pv[i].y, pscale[i], sv[i].y * sscale[i]);
        o.z = fmaf(pv[i].z, pscale[i], sv[i].z * sscale[i]);
        o.w = fmaf(pv[i].w, pscale[i], sv[i].w * sscale[i]);
        __builtin_nontemporal_store(o, (vfloat4*)(out + base));
    }
}

extern "C" void kernel_launch(void* const* d_in, const int* in_sizes, int n_in,
                              void* d_out, int out_size, void* d_ws, size_t ws_size,
                              hipStream_t stream) {
    // setup_inputs() order: prefix_output, prefix_lse, suffix_output, suffix_lse
    const float* p_out = (const float*)d_in[0];
    const float* p_lse = (const float*)d_in[1];
    const float* s_out = (const float*)d_in[2];
    const float* s_lse = (const float*)d_in[3];

    float* out     = (float*)d_out;
    float* out_lse = out + (size_t)NUM_TOKENS * NUM_HEADS * HEAD_SIZE;

    constexpr int total_pairs = NUM_TOKENS * NUM_HEADS;            // 524288
    constexpr int total_waves = total_pairs / ILP;                 // 131072
    constexpr int blocks      = total_waves / WAVES_PER_BLOCK;     // 16384

    merge_attn_states_kernel<<<blocks, 256, 0, stream>>>(
        p_out, p_lse, s_out, s_lse, out, out_lse);
}